// RegularizedFMNet_19207093748110
// MI455X (gfx1250) — compile-verified
//
#include <hip/hip_runtime.h>
#include <math.h>

// Problem constants (match reference).
static constexpr int Bn = 8;
static constexpr int Vn = 5000;
static constexpr int Cn = 256;
static constexpr int Kn = 200;

typedef float v2f __attribute__((ext_vector_type(2)));
typedef float v8f __attribute__((ext_vector_type(8)));

// D = A(16x4 f32) * B(4x16 f32) + C(16x16 f32), full f32 precision.
#define WMMA_F32_16x16x4(a, b, c) \
  __builtin_amdgcn_wmma_f32_16x16x4_f32(false, (a), false, (b), (short)0, (c), false, false)

// ---------------------------------------------------------------------------
// Scalars: gamma = sigmoid(gamma_raw); lambda = clip(10 + softplus(l)*990/1000)
// ---------------------------------------------------------------------------
__global__ void scalars_kernel(const float* __restrict__ graw,
                               const float* __restrict__ lraw,
                               float* __restrict__ scal) {
  float g = 1.0f / (1.0f + expf(-graw[0]));
  float x = lraw[0];
  float sp = (x > 20.0f) ? x : log1pf(expf(x));
  float l = 10.0f + sp * (1000.0f - 10.0f) / 1000.0f;
  l = fminf(fmaxf(l, 10.0f), 1000.0f);
  scal[0] = g;
  scal[1] = l;
}

// ---------------------------------------------------------------------------
// Projection GEMM: Out(K x C) = E(K x V) * F(V x C), batched over blockIdx.z.
// One wave per 16(M) x 64(N) tile; reduction over V in steps of 4 via
// v_wmma_f32_16x16x4_f32. Row index clamped (K=200 is not a multiple of 16)
// so EXEC stays all-ones for WMMA; stores are guarded.
// ---------------------------------------------------------------------------
__global__ __launch_bounds__(32)
void proj_gemm_kernel(const float* __restrict__ E,
                      const float* __restrict__ F,
                      float* __restrict__ Out) {
  const int lane = threadIdx.x;   // 0..31
  const int half = lane >> 4;     // 0/1 (selects K pair)
  const int l16  = lane & 15;
  const int b  = blockIdx.z;
  const int mt = blockIdx.y;      // 16-row tile of K
  const int nt = blockIdx.x;      // 64-col tile of C

  const float* Eb = E + (size_t)b * Kn * Vn;
  const float* Fb = F + (size_t)b * Vn * Cn;
  float*       Ob = Out + (size_t)b * Kn * Cn;

  const int mrow   = mt * 16 + l16;
  const int mclamp = (mrow < Kn) ? mrow : (Kn - 1);
  const float* Erow = Eb + (size_t)mclamp * Vn;
  const int col0 = nt * 64 + l16;

  v8f acc0 = {}, acc1 = {}, acc2 = {}, acc3 = {};
  for (int v = 0; v < Vn; v += 4) {
    const int kk = v + 2 * half;
    v2f a;  a.x  = Erow[kk];      a.y  = Erow[kk + 1];
    const float* F0 = Fb + (size_t)kk * Cn + col0;
    const float* F1 = F0 + Cn;
    v2f b0; b0.x = F0[0];   b0.y = F1[0];
    v2f b1; b1.x = F0[16];  b1.y = F1[16];
    v2f b2; b2.x = F0[32];  b2.y = F1[32];
    v2f b3; b3.x = F0[48];  b3.y = F1[48];
    acc0 = WMMA_F32_16x16x4(a, b0, acc0);
    acc1 = WMMA_F32_16x16x4(a, b1, acc1);
    acc2 = WMMA_F32_16x16x4(a, b2, acc2);
    acc3 = WMMA_F32_16x16x4(a, b3, acc3);
  }

  // C/D layout: VGPR r holds rows (mt*16 + r + 8*half), col = tile + (lane&15).
  const int rbase = mt * 16 + 8 * half;
  for (int r = 0; r < 8; ++r) {
    const int row = rbase + r;
    if (row < Kn) {
      float* Orow = Ob + (size_t)row * Cn + nt * 64 + l16;
      Orow[0]  = acc0[r];
      Orow[16] = acc1[r];
      Orow[32] = acc2[r];
      Orow[48] = acc3[r];
    }
  }
}

// ---------------------------------------------------------------------------
// Gram GEMM: Out(K x K) = L(K x C) * R(K x C)^T. Because the right operand is
// transposed, its WMMA B-fragment is simply an A-layout load of R.
// One wave per 16x16 tile.
// ---------------------------------------------------------------------------
__global__ __launch_bounds__(32)
void gram_gemm_kernel(const float* __restrict__ L,
                      const float* __restrict__ R,
                      float* __restrict__ Out) {
  const int lane = threadIdx.x;
  const int half = lane >> 4;
  const int l16  = lane & 15;
  const int b  = blockIdx.z;
  const int mt = blockIdx.y;
  const int nt = blockIdx.x;

  const float* Lb = L + (size_t)b * Kn * Cn;
  const float* Rb = R + (size_t)b * Kn * Cn;

  const int mrow = mt * 16 + l16;
  const int nrow = nt * 16 + l16;
  const int mc = (mrow < Kn) ? mrow : (Kn - 1);
  const int nc = (nrow < Kn) ? nrow : (Kn - 1);
  const float* Lrow = Lb + (size_t)mc * Cn;
  const float* Rrow = Rb + (size_t)nc * Cn;

  v8f acc = {};
  for (int c = 0; c < Cn; c += 4) {
    const int kk = c + 2 * half;
    v2f a;  a.x  = Lrow[kk]; a.y  = Lrow[kk + 1];
    v2f bb; bb.x = Rrow[kk]; bb.y = Rrow[kk + 1];
    acc = WMMA_F32_16x16x4(a, bb, acc);
  }

  float* Ob = Out + (size_t)b * Kn * Kn;
  const int rbase = mt * 16 + 8 * half;
  const int col = nt * 16 + l16;
  for (int r = 0; r < 8; ++r) {
    const int row = rbase + r;
    if (row < Kn && col < Kn) Ob[(size_t)row * Kn + col] = acc[r];
  }
}

// ---------------------------------------------------------------------------
// Resolvent mask D[b,i,j] (8 x 200 x 200). One 256-thread block per batch.
// ---------------------------------------------------------------------------
__global__ void mask_kernel(const float* __restrict__ evx,
                            const float* __restrict__ evy,
                            const float* __restrict__ scal,
                            float* __restrict__ D) {
  __shared__ float red[256];
  __shared__ float gx[Kn];
  __shared__ float gy[Kn];
  const int b = blockIdx.x;
  const int t = threadIdx.x;
  const float* ex = evx + (size_t)b * Kn;
  const float* ey = evy + (size_t)b * Kn;

  float m = -1e30f;
  for (int i = t; i < Kn; i += 256) {
    m = fmaxf(m, ex[i]);
    m = fmaxf(m, ey[i]);
  }
  red[t] = m;
  __syncthreads();
  for (int s = 128; s > 0; s >>= 1) {
    if (t < s) red[t] = fmaxf(red[t], red[t + s]);
    __syncthreads();
  }
  const float scaling = red[0];
  const float gamma = scal[0];

  for (int i = t; i < Kn; i += 256) {
    gx[i] = powf(ex[i] / scaling, gamma);
    gy[i] = powf(ey[i] / scaling, gamma);
  }
  __syncthreads();

  float* Db = D + (size_t)b * Kn * Kn;
  for (int idx = t; idx < Kn * Kn; idx += 256) {
    const int i = idx / Kn;
    const int j = idx - i * Kn;
    const float g1 = gx[i];
    const float g2 = gy[j];
    const float d1 = 1.0f / (g1 * g1 + 1.0f);
    const float d2 = 1.0f / (g2 * g2 + 1.0f);
    const float re = g2 * d2 - g1 * d1;
    const float im = d2 - d1;
    Db[idx] = re * re + im * im;
  }
}

// ---------------------------------------------------------------------------
// Solve kernel: one workgroup per (b,i). Builds M = AAt[b] + lambda*diag(D[b,i,:])
// augmented with rhs = BAt[b,i,:] in dynamic LDS (200 x 201 floats = 161.6 KB;
// only legal on CDNA5's 320 KB WGP LDS), then unpivoted Gaussian elimination
// (M is SPD) and a parallel Jordan back-substitution.
// ---------------------------------------------------------------------------
__global__ void solve_kernel(const float* __restrict__ AAt,
                             const float* __restrict__ BAt,
                             const float* __restrict__ Dmask,
                             const float* __restrict__ scal,
                             float* __restrict__ Out) {
  extern __shared__ float sh[];
  const int LDM = Kn + 1;              // 201: row = 200 matrix cols + rhs
  float* Mloc = sh;                    // Kn * LDM
  float* x    = sh + Kn * LDM;         // Kn

  const int b = blockIdx.y;
  const int i = blockIdx.x;
  const int t = threadIdx.x;
  const int nt = blockDim.x;

  const float lmbda = scal[1];
  const float* Ab   = AAt   + (size_t)b * Kn * Kn;
  const float* rhs  = BAt   + ((size_t)b * Kn + i) * Kn;
  const float* Drow = Dmask + ((size_t)b * Kn + i) * Kn;

  for (int idx = t; idx < Kn * Kn; idx += nt) {
    const int r = idx / Kn;
    const int c = idx - r * Kn;
    Mloc[r * LDM + c] = Ab[idx];
  }
  for (int r = t; r < Kn; r += nt) Mloc[r * LDM + Kn] = rhs[r];
  __syncthreads();
  for (int j = t; j < Kn; j += nt) Mloc[j * LDM + j] += lmbda * Drow[j];
  __syncthreads();

  // Forward elimination (no pivoting: M is symmetric positive definite).
  for (int k = 0; k < Kn - 1; ++k) {
    const float pivinv = 1.0f / Mloc[k * LDM + k];
    const float* prow = &Mloc[k * LDM];
    for (int r = k + 1 + t; r < Kn; r += nt) {
      float* row = &Mloc[r * LDM];
      const float f = row[k] * pivinv;
      for (int c = k + 1; c <= Kn; ++c) row[c] -= f * prow[c];
    }
    __syncthreads();
  }

  // Back substitution, parallel rank-1 rhs updates.
  for (int k = Kn - 1; k >= 0; --k) {
    if (t == 0) x[k] = Mloc[k * LDM + Kn] / Mloc[k * LDM + k];
    __syncthreads();
    const float xk = x[k];
    for (int r = t; r < k; r += nt) Mloc[r * LDM + Kn] -= Mloc[r * LDM + k] * xk;
    __syncthreads();
  }

  float* Orow = Out + ((size_t)b * Kn + i) * Kn;
  for (int r = t; r < Kn; r += nt) Orow[r] = x[r];
}

// ---------------------------------------------------------------------------
// Host launcher.
// ---------------------------------------------------------------------------
extern "C" void kernel_launch(void* const* d_in, const int* in_sizes, int n_in,
                              void* d_out, int out_size, void* d_ws, size_t ws_size,
                              hipStream_t stream) {
  const float* feat_x  = (const float*)d_in[0];
  const float* feat_y  = (const float*)d_in[1];
  const float* evals_x = (const float*)d_in[2];
  const float* evals_y = (const float*)d_in[3];
  const float* etx     = (const float*)d_in[4];
  const float* ety     = (const float*)d_in[5];
  const float* graw    = (const float*)d_in[6];
  const float* lraw    = (const float*)d_in[7];
  float* out = (float*)d_out;

  // Workspace layout (floats): scalars | A | B | AAt | BAt | D
  float* ws   = (float*)d_ws;
  float* scal = ws;                                  // 16 floats
  float* Aw   = ws + 16;                             // B*K*C
  float* Bw   = Aw + (size_t)Bn * Kn * Cn;           // B*K*C
  float* AAt  = Bw + (size_t)Bn * Kn * Cn;           // B*K*K
  float* BAt  = AAt + (size_t)Bn * Kn * Kn;          // B*K*K
  float* Dw   = BAt + (size_t)Bn * Kn * Kn;          // B*K*K

  scalars_kernel<<<1, 1, 0, stream>>>(graw, lraw, scal);

  const dim3 pg(Cn / 64, (Kn + 15) / 16, Bn);        // (4, 13, 8)
  proj_gemm_kernel<<<pg, 32, 0, stream>>>(etx, feat_x, Aw);
  proj_gemm_kernel<<<pg, 32, 0, stream>>>(ety, feat_y, Bw);

  const dim3 gg((Kn + 15) / 16, (Kn + 15) / 16, Bn); // (13, 13, 8)
  gram_gemm_kernel<<<gg, 32, 0, stream>>>(Aw, Aw, AAt);
  gram_gemm_kernel<<<gg, 32, 0, stream>>>(Bw, Aw, BAt);

  mask_kernel<<<Bn, 256, 0, stream>>>(evals_x, evals_y, scal, Dw);

  const size_t smem = (size_t)(Kn * (Kn + 1) + Kn) * sizeof(float); // 161,600 B
  (void)hipFuncSetAttribute((const void*)solve_kernel,
                            hipFuncAttributeMaxDynamicSharedMemorySize, (int)smem);
  solve_kernel<<<dim3(Kn, Bn), 256, smem, stream>>>(AAt, BAt, Dw, scal, out);
}